// Patcher_12034498363986
// MI455X (gfx1250) — compile-verified
//
#include <hip/hip_runtime.h>

// Problem constants (match reference: B=64, T=512, N=1024, MAX_TIME_F=1, MAX_SPACE_F=32)
#define Bc 64
#define Tc 512
#define Nc 1024
#define SF 32                    // MAX_SPACE_F
#define nSc (Nc / SF)            // 32 spatial patches
#define S2c (nSc + 1)            // +1 CLS column
#define PATCH_ELEMS (Bc * Tc * Nc)      // 33,554,432 floats
#define MASK_ELEMS  (Bc * Tc * S2c)     // 1,081,344 per mask/stamp tensor

typedef float __attribute__((ext_vector_type(4))) f4;

// ---------------------------------------------------------------------------
// Kernel 1: streaming patch copy + boundary-patch blend.
// grid = (T*N/4/256, B), block = 256.  b is block-uniform -> scalar descriptor
// load for pad_space_len.  B128 (float4) NT loads/stores: pure HBM bandwidth.
// ---------------------------------------------------------------------------
__global__ __launch_bounds__(256) void patch_copy_kernel(
    const float* __restrict__ spikes,
    const int*   __restrict__ pad_space_len,
    float*       __restrict__ out)
{
    const int b = blockIdx.y;
    const unsigned rem = blockIdx.x * 256u + threadIdx.x;   // float4 idx within batch [0, T*N/4)

    const int psv  = pad_space_len[b];
    const int psl  = psv & (SF - 1);     // leftover valid channels in boundary patch
    const int pidx = psv >> 5;           // boundary patch index

    const unsigned col4 = rem & (Nc / 4 - 1);   // float4 column within a row: 0..255
    const int s = (int)(col4 >> 3);             // spatial patch 0..31
    const int c = (int)(col4 & 7) << 2;         // element offset within patch: 0,4,...,28

    const unsigned gidx = (unsigned)b * (Tc * Nc / 4) + rem;  // global float4 index
    const f4* src = (const f4*)spikes + gidx;

    f4 v = __builtin_nontemporal_load(src);

    if (psl != 0 && s == pidx) {
        // blend: pos < psl -> current patch, pos >= psl -> previous patch (32 floats back)
        f4 p = __builtin_nontemporal_load(src - (SF / 4));
        f4 r;
        r.x = (c + 0 < psl) ? v.x : p.x;
        r.y = (c + 1 < psl) ? v.y : p.y;
        r.z = (c + 2 < psl) ? v.z : p.z;
        r.w = (c + 3 < psl) ? v.w : p.w;
        v = r;
    }
    __builtin_nontemporal_store(v, (f4*)out + gidx);
}

// ---------------------------------------------------------------------------
// Kernel 2: per-(b,s) OR-reduction of space_attn_mask over 32 channels.
// One wave32 per batch (64 blocks x 32 threads) -> s_valid [B, nS] in d_ws.
// ---------------------------------------------------------------------------
__global__ void svalid_kernel(const int* __restrict__ space_mask,
                              float*     __restrict__ sv)
{
    const int b = blockIdx.x;
    const int s = threadIdx.x;           // 0..31
    const int* p = space_mask + b * Nc + s * SF;
    int acc = 0;
    #pragma unroll
    for (int c = 0; c < SF; ++c) acc |= p[c];   // mask values are 0/1: sum>0 <=> OR!=0
    sv[b * nSc + s] = (acc != 0) ? 1.0f : 0.0f;
}

// ---------------------------------------------------------------------------
// Kernel 3: smask / tmask / spacestamps / timestamps, 4 outputs per thread.
// With MAX_TIME_F==1 the time window reduce is just the mask value itself.
// ---------------------------------------------------------------------------
__global__ __launch_bounds__(256) void mask_stamp_kernel(
    const int*   __restrict__ time_mask,   // [B, T] 0/1
    const float* __restrict__ s_valid,     // [B, nS]
    float*       __restrict__ out)
{
    const unsigned idx = blockIdx.x * 256u + threadIdx.x;
    if (idx >= (unsigned)MASK_ELEMS) return;

    const unsigned b   = idx / (Tc * S2c);
    const unsigned rem = idx - b * (Tc * S2c);
    const unsigned t   = rem / S2c;
    const unsigned j   = rem - t * S2c;    // 0 = CLS column

    const float sm = (j == 0) ? 1.0f : s_valid[b * nSc + (j - 1)];
    const float tm = (j == 0) ? 1.0f : (time_mask[b * Tc + t] != 0 ? 1.0f : 0.0f);

    float* base = out + PATCH_ELEMS;       // outputs concatenated in return order
    base[idx]                     = sm;         // smask
    base[1u * MASK_ELEMS + idx]   = tm;         // tmask
    base[2u * MASK_ELEMS + idx]   = (float)j;   // spacestamps
    base[3u * MASK_ELEMS + idx]   = (float)t;   // timestamps
}

extern "C" void kernel_launch(void* const* d_in, const int* in_sizes, int n_in,
                              void* d_out, int out_size, void* d_ws, size_t ws_size,
                              hipStream_t stream)
{
    (void)in_sizes; (void)n_in; (void)out_size; (void)ws_size;

    const float* spikes        = (const float*)d_in[0];
    const int*   pad_space_len = (const int*)  d_in[1];
    // d_in[2] = pad_time_len: unused by the forward pass
    const int*   time_mask     = (const int*)  d_in[3];
    const int*   space_mask    = (const int*)  d_in[4];

    float* out = (float*)d_out;
    float* sv  = (float*)d_ws;   // 64*32 floats of scratch

    // Bulk patch copy: 512 x 64 blocks of 256 threads, one float4 per thread.
    patch_copy_kernel<<<dim3(Tc * Nc / 4 / 256, Bc), 256, 0, stream>>>(
        spikes, pad_space_len, out);

    // s_valid reduction, then masks/stamps (same stream -> ordered).
    svalid_kernel<<<Bc, nSc, 0, stream>>>(space_mask, sv);
    mask_stamp_kernel<<<(MASK_ELEMS + 255) / 256, 256, 0, stream>>>(
        time_mask, sv, out);
}